// Sample_11690900979980
// MI455X (gfx1250) — compile-verified
//
#include <hip/hip_runtime.h>
#include <cstdint>

// Problem constants (from reference): B=8, N=131072, m=1024, points f32 [B,N,3].
#define BATCHES 8
#define NPTS    131072
#define MSEL    1024
#define GPB     8                 // blocks (WGPs) per batch
#define TPB     1024              // threads per block (32 wave32 waves)
#define PPB     (NPTS / GPB)      // 16384 points per block
#define PPT     (PPB / TPB)       // 16 points per thread (min-dist lives in VGPRs)

struct Cand { float v; int i; float x, y, z; int pad[3]; };   // 32 B

__global__ void fps_init_kernel(int* __restrict__ cnt) {
    if (threadIdx.x < BATCHES) cnt[threadIdx.x] = 0;
}

__launch_bounds__(TPB)
__global__ void fps_kernel(const float* __restrict__ points,
                           float* __restrict__ out,
                           int* __restrict__ cnt,
                           Cand* __restrict__ cands) {
    // 192 KB of LDS for this block's points — only possible on CDNA5 (320 KB/WGP).
    __shared__ alignas(16) float pts[PPB * 3];
    __shared__ float sval[32];
    __shared__ int   sidx[32];
    __shared__ float qsh[3];

    const int tid = threadIdx.x;
    const int g   = blockIdx.x;          // block within batch
    const int b   = blockIdx.y;          // batch
    const int baseLocal = g * PPB;       // first point index (within batch) owned here

    // ---- Stage 192 KB of points into LDS with CDNA5 async global->LDS copies ----
    {
        const uint8_t* gbase = (const uint8_t*)points
                             + ((size_t)b * NPTS + (size_t)baseLocal) * 3u * sizeof(float);
        uint32_t lds0 = (uint32_t)(uintptr_t)&pts[0];   // low 32 bits of LDS aperture = LDS offset
        const int n16 = PPB * 3 * 4 / 16;               // 12288 x b128 lane-ops
        for (int i = tid; i < n16; i += TPB) {
            uint32_t ldsoff = lds0 + (uint32_t)i * 16u;
            uint64_t gaddr  = (uint64_t)(uintptr_t)(gbase + (size_t)i * 16u);
            asm volatile("global_load_async_to_lds_b128 %0, %1, off"
                         :: "v"(ldsoff), "v"(gaddr) : "memory");
        }
        asm volatile("s_wait_asynccnt 0" ::: "memory");
    }
    __syncthreads();

    // Initial query point = point 0 of the batch (FPS seed index 0).
    const float* p0 = points + (size_t)b * NPTS * 3;
    float qx = p0[0], qy = p0[1], qz = p0[2];

    if (g == 0 && tid == 0) {
        out[(size_t)b * MSEL] = 0.0f;                        // idx[b][0] = 0
        float* oxyz = out + (size_t)BATCHES * MSEL;
        size_t o = (size_t)b * MSEL * 3;
        oxyz[o + 0] = qx; oxyz[o + 1] = qy; oxyz[o + 2] = qz;
    }

    float mind[PPT];
    #pragma unroll
    for (int k = 0; k < PPT; ++k) mind[k] = 1e10f;

    int* mycnt = cnt + b;
    // Candidate slots double-buffered by iteration parity -> single barrier/iter.
    volatile Cand* cbuf[2];
    cbuf[0] = (volatile Cand*)(cands + (b * 2 + 0) * GPB);
    cbuf[1] = (volatile Cand*)(cands + (b * 2 + 1) * GPB);

    for (int iter = 1; iter < MSEL; ++iter) {
        // ---- distances to q, running min, thread-local argmax ----
        float bv = -1.0f; int bi = 0x7fffffff;
        #pragma unroll
        for (int k = 0; k < PPT; ++k) {
            int j = k * TPB + tid;                 // ascending => first-max tie-break
            float x = pts[3 * j + 0];
            float y = pts[3 * j + 1];
            float z = pts[3 * j + 2];
            float dx = x - qx, dy = y - qy, dz = z - qz;
            float d  = fmaf(dz, dz, fmaf(dy, dy, dx * dx));
            float md = fminf(mind[k], d);
            mind[k] = md;
            if (md > bv) { bv = md; bi = baseLocal + j; }
        }
        // ---- wave32 shuffle argmax (lowest index wins ties, like jnp.argmax) ----
        #pragma unroll
        for (int off = 16; off > 0; off >>= 1) {
            float ov = __shfl_xor(bv, off, 32);
            int   oi = __shfl_xor(bi, off, 32);
            if (ov > bv || (ov == bv && oi < bi)) { bv = ov; bi = oi; }
        }
        int wave = tid >> 5, lane = tid & 31;
        if (lane == 0) { sval[wave] = bv; sidx[wave] = bi; }
        __syncthreads();

        if (tid < 32) {
            bv = sval[tid]; bi = sidx[tid];
            #pragma unroll
            for (int off = 16; off > 0; off >>= 1) {
                float ov = __shfl_xor(bv, off, 32);
                int   oi = __shfl_xor(bi, off, 32);
                if (ov > bv || (ov == bv && oi < bi)) { bv = ov; bi = oi; }
            }
            if (tid == 0) {
                volatile Cand* vc = cbuf[iter & 1];
                // publish this block's candidate (value, index, coords)
                int j = bi - baseLocal;
                vc[g].v = bv; vc[g].i = bi;
                vc[g].x = pts[3 * j + 0];
                vc[g].y = pts[3 * j + 1];
                vc[g].z = pts[3 * j + 2];
                __threadfence();
                atomicAdd(mycnt, 1);
                // single barrier per iteration: counter reaches GPB*iter when
                // every block of this batch has published for this iteration
                int target = GPB * iter;
                while (__hip_atomic_load(mycnt, __ATOMIC_RELAXED,
                                         __HIP_MEMORY_SCOPE_AGENT) < target)
                    __builtin_amdgcn_s_sleep(1);
                __threadfence();
                // reduce the GPB candidates -> batch-global winner
                float wv = -1.0f; int wi = 0x7fffffff;
                float wx = 0.f, wy = 0.f, wz = 0.f;
                for (int q = 0; q < GPB; ++q) {
                    float v = vc[q].v; int i2 = vc[q].i;
                    if (v > wv || (v == wv && i2 < wi)) {
                        wv = v; wi = i2;
                        wx = vc[q].x; wy = vc[q].y; wz = vc[q].z;
                    }
                }
                if (g == 0) {
                    out[(size_t)b * MSEL + iter] = (float)wi;
                    float* oxyz = out + (size_t)BATCHES * MSEL;
                    size_t o = ((size_t)b * MSEL + (size_t)iter) * 3;
                    oxyz[o + 0] = wx; oxyz[o + 1] = wy; oxyz[o + 2] = wz;
                }
                qsh[0] = wx; qsh[1] = wy; qsh[2] = wz;
            }
        }
        __syncthreads();
        qx = qsh[0]; qy = qsh[1]; qz = qsh[2];
    }
}

extern "C" void kernel_launch(void* const* d_in, const int* in_sizes, int n_in,
                              void* d_out, int out_size, void* d_ws, size_t ws_size,
                              hipStream_t stream) {
    const float* points = (const float*)d_in[0];
    float* out = (float*)d_out;

    int*  cnt   = (int*)d_ws;                       // 8 monotonic barrier counters
    Cand* cands = (Cand*)((char*)d_ws + 256);       // 8 batches x 2 parities x 8 blocks x 32 B

    fps_init_kernel<<<1, 64, 0, stream>>>(cnt);     // reset counters every call (replay-safe)

    dim3 grid(GPB, BATCHES);
    fps_kernel<<<grid, TPB, 0, stream>>>(points, out, cnt, cands);
}